// NoOffRoad_38019050504607
// MI455X (gfx1250) — compile-verified
//
#include <hip/hip_runtime.h>
#include <hip/hip_bf16.h>
#include <math.h>

#define N_QUERY 8192
#define N_RG    40000
#define NCHUNK  100           // rg dimension split into 100 chunks
#define TILES_PER_CHUNK 25    // 25 tiles * 16 cols = 400 pts/chunk ; 100*400 = 40000 exactly
#define QTILES  4             // query tiles (of 16) per wave -> 4 WMMAs per B tile
#define MASK_BIG 1.0e30f

typedef __attribute__((ext_vector_type(2))) float v2f;
typedef __attribute__((ext_vector_type(8))) float v8f;

// ---------------------------------------------------------------------------
// Kernel 1: WMMA-tiled masked-distance argmin.
// One wave handles 64 queries (4 A-tiles) x 400 rg points.
// D = q^2 + p^2 - 2 q.p  (exact masked squared distance, K=4):
//   A row m  = [qx, qy, 1, q^2]
//   B col n  = [-2px, -2py, p2_or_mask, 1]
// Branchless inner loop (EXEC all-1s for WMMA) + 2-stage load pipeline.
// ---------------------------------------------------------------------------
__global__ __launch_bounds__(32)
void argmin_tiles_kernel(const float* __restrict__ traj,
                         const float* __restrict__ rg_xyz,
                         const int*   __restrict__ rg_type,
                         float* __restrict__ partmin,
                         int*   __restrict__ partidx)
{
    const int lane = threadIdx.x;     // 0..31
    const int half = lane >> 4;       // 0: K=0,1 / D rows 0..7 ; 1: K=2,3 / D rows 8..15
    const int l15  = lane & 15;
    const int qbase = blockIdx.x * (16 * QTILES);
    const int chunk = blockIdx.y;

    // A matrices 16x4 (f32): lanes 0-15 hold (K=0,K=1) of row M=l15,
    //                        lanes 16-31 hold (K=2,K=3) of row M=l15.
    v2f A[QTILES];
#pragma unroll
    for (int s = 0; s < QTILES; ++s) {
        const int q = qbase + s * 16 + l15;
        const float qx = traj[2 * q + 0];
        const float qy = traj[2 * q + 1];
        const float q2 = fmaf(qx, qx, qy * qy);
        A[s].x = half ? 1.0f : qx;    // K=2: multiplies (p^2 | mask)
        A[s].y = half ? q2   : qy;    // K=3: multiplies 1 -> adds q^2
    }

    float minv[QTILES][8];
    int   mini[QTILES][8];
#pragma unroll
    for (int s = 0; s < QTILES; ++s)
#pragma unroll
        for (int v = 0; v < 8; ++v) { minv[s][v] = 3.0e38f; mini[s][v] = 0; }

    // Process one 16-column B tile: build B branchlessly, 4 WMMAs, min-update.
    auto process_tile = [&](float x, float y, int ty, int base_n) {
        const bool  edge = (ty == 15) || (ty == 16);
        const float p2m  = edge ? fmaf(x, x, y * y) : MASK_BIG;
        v2f B;
        B.x = half ? p2m  : (-2.0f * x);   // K=0 / K=2 rows
        B.y = half ? 1.0f : (-2.0f * y);   // K=1 / K=3 rows
#pragma unroll
        for (int s = 0; s < QTILES; ++s) {
            v8f Cz = {};
            const v8f D = __builtin_amdgcn_wmma_f32_16x16x4_f32(
                              false, A[s], false, B, (short)0, Cz, false, false);
            // Lane owns column N = base_n, rows M = v + 8*half. Strict '<'
            // keeps the first (lowest-index) minimum since n scans ascending.
#pragma unroll
            for (int v = 0; v < 8; ++v) {
                const float d = D[v];
                if (d < minv[s][v]) { minv[s][v] = d; mini[s][v] = base_n; }
            }
        }
    };

    // Per-lane stepped pointers + 2-stage software pipeline (last tile peeled
    // so the prefetch never reads past N_RG).
    const int n0 = chunk * TILES_PER_CHUNK * 16 + l15;
    const float* __restrict__ pxy = rg_xyz + 3 * (size_t)n0;
    const int*   __restrict__ pty = rg_type + n0;

    float cx  = pxy[0];
    float cy  = pxy[1];
    int   cty = pty[0];
    int   cn  = n0;

    for (int t = 0; t < TILES_PER_CHUNK - 1; ++t) {
        pxy += 48;                    // 16 points * 3 floats
        pty += 16;
        const float nx_ = pxy[0];     // prefetch next tile
        const float ny_ = pxy[1];
        const int   nty = pty[0];

        process_tile(cx, cy, cty, cn);

        cx = nx_; cy = ny_; cty = nty; cn += 16;
    }
    process_tile(cx, cy, cty, cn);    // peeled last tile

    // Column reduction: rows 0..7 live in lanes 0-15, rows 8..15 in lanes 16-31.
    // xor offsets <= 8 stay inside each 16-lane half.
#pragma unroll
    for (int s = 0; s < QTILES; ++s) {
#pragma unroll
        for (int off = 8; off > 0; off >>= 1) {
#pragma unroll
            for (int v = 0; v < 8; ++v) {
                const float ov = __shfl_xor(minv[s][v], off, 32);
                const int   oi = __shfl_xor(mini[s][v], off, 32);
                if (ov < minv[s][v] || (ov == minv[s][v] && oi < mini[s][v])) {
                    minv[s][v] = ov; mini[s][v] = oi;
                }
            }
        }
#pragma unroll
        for (int v = 0; v < 8; ++v) {
            if (l15 == v) {
                const int q = qbase + s * 16 + v + 8 * half;
                partmin[q * NCHUNK + chunk] = minv[s][v];
                partidx[q * NCHUNK + chunk] = mini[s][v];
            }
        }
    }
}

// ---------------------------------------------------------------------------
// Kernel 2: reduce per-query partials, signed-distance epilogue, global sums.
// ---------------------------------------------------------------------------
__global__ __launch_bounds__(256)
void finalize_kernel(const float* __restrict__ traj,
                     const float* __restrict__ rg_xyz,
                     const float* __restrict__ rg_dir,
                     const int*   __restrict__ rg_ids,
                     const float* __restrict__ partmin,
                     const int*   __restrict__ partidx,
                     float* __restrict__ acc)
{
    const int q = blockIdx.x * blockDim.x + threadIdx.x;
    float a = 0.0f, cnt = 0.0f;

    if (q < N_QUERY) {
        float bestv = 3.0e38f;
        int   besti = 0x7fffffff;
        for (int c = 0; c < NCHUNK; ++c) {
            const float v = partmin[q * NCHUNK + c];
            const int   i = partidx[q * NCHUNK + c];
            if (v < bestv || (v == bestv && i < besti)) { bestv = v; besti = i; }
        }
        const int nearest = besti;
        const int prior   = (nearest > 0) ? nearest - 1 : 0;

        const float qx = traj[2 * q + 0];
        const float qy = traj[2 * q + 1];
        const float ex = qx - rg_xyz[3 * nearest + 0];
        const float ey = qy - rg_xyz[3 * nearest + 1];
        const float dnx = rg_dir[3 * nearest + 0];
        const float dny = rg_dir[3 * nearest + 1];
        const float dpx = rg_dir[3 * prior + 0];
        const float dpy = rg_dir[3 * prior + 1];

        const float cross_n = ex * dny - ey * dnx;
        const float cross_p = ex * dpy - ey * dpx;
        const bool  same    = (rg_ids[nearest] == rg_ids[prior]);
        const float cross   = (same && (cross_p < cross_n)) ? cross_p : cross_n;
        const float sgn     = (cross > 0.0f) ? 1.0f : ((cross < 0.0f) ? -1.0f : 0.0f);
        const float sd      = sqrtf(ex * ex + ey * ey) * sgn;

        a   = fmaxf(1.0f + sd, 0.0f);
        cnt = (a > 0.0f) ? 1.0f : 0.0f;
    }

    // wave32 reduction, then one atomic per wave
#pragma unroll
    for (int off = 16; off > 0; off >>= 1) {
        a   += __shfl_down(a,   off, 32);
        cnt += __shfl_down(cnt, off, 32);
    }
    if ((threadIdx.x & 31) == 0) {
        atomicAdd(&acc[0], a);
        atomicAdd(&acc[1], cnt);
    }
}

__global__ void init_acc_kernel(float* acc) {
    acc[0] = 0.0f;
    acc[1] = 0.0f;
}

__global__ void writeout_kernel(const float* __restrict__ acc,
                                float* __restrict__ out) {
    out[0] = acc[0] / (acc[1] + 1.0e-6f);
}

// ---------------------------------------------------------------------------
extern "C" void kernel_launch(void* const* d_in, const int* in_sizes, int n_in,
                              void* d_out, int out_size, void* d_ws, size_t ws_size,
                              hipStream_t stream) {
    const float* traj    = (const float*)d_in[0];   // (8192, 2)  f32
    const float* rg_xyz  = (const float*)d_in[1];   // (40000, 3) f32
    const float* rg_dir  = (const float*)d_in[2];   // (40000, 3) f32
    const int*   rg_ids  = (const int*)d_in[3];     // (40000,)   i32
    const int*   rg_type = (const int*)d_in[4];     // (40000,)   i32
    float* out = (float*)d_out;

    (void)in_sizes; (void)n_in; (void)out_size; (void)ws_size;

    // Workspace: [acc: 2 f32][pad to 256B][partmin: 8192*100 f32][partidx: 8192*100 i32]
    char*  ws      = (char*)d_ws;
    float* acc     = (float*)ws;
    float* partmin = (float*)(ws + 256);
    int*   partidx = (int*)(ws + 256 + sizeof(float) * (size_t)N_QUERY * NCHUNK);

    init_acc_kernel<<<1, 1, 0, stream>>>(acc);

    dim3 grid1(N_QUERY / (16 * QTILES), NCHUNK);
    argmin_tiles_kernel<<<grid1, 32, 0, stream>>>(traj, rg_xyz, rg_type,
                                                  partmin, partidx);

    finalize_kernel<<<N_QUERY / 256, 256, 0, stream>>>(traj, rg_xyz, rg_dir, rg_ids,
                                                       partmin, partidx, acc);

    writeout_kernel<<<1, 1, 0, stream>>>(acc, out);
}